// LBPModule_19181323944545
// MI455X (gfx1250) — compile-verified
//
#include <hip/hip_runtime.h>

typedef __attribute__((ext_vector_type(2))) float v2f;
typedef __attribute__((ext_vector_type(8))) float v8f;

// ---------------- workspace layout (in floats) ----------------
constexpr size_t N_P     = 16u * 256u * 256u;        // 1,048,576
constexpr size_t N_LBP   = 16u * 8u * 256u * 256u;   // 8,388,608
constexpr size_t N_S     = 16u * 4u * 64u * 56u;     // 229,376
constexpr size_t N_Z     = 16u * 16u * 64u * 56u;    // 917,504
constexpr size_t OFF_P    = 0;
constexpr size_t OFF_LBP  = OFF_P + N_P;
constexpr size_t OFF_MN   = OFF_LBP + N_LBP;   // 128
constexpr size_t OFF_MX   = OFF_MN + 128;      // 128
constexpr size_t OFF_HIST = OFF_MX + 128;      // 64*64
constexpr size_t OFF_TOT  = OFF_HIST + 4096;   // 1
constexpr size_t OFF_S    = OFF_TOT + 1;
constexpr size_t OFF_MEAN = OFF_S + N_S;       // 64
constexpr size_t OFF_ISTD = OFF_MEAN + 64;     // 64
constexpr size_t OFF_Z    = OFF_ISTD + 64;

__device__ __forceinline__ float leaky(float v) { return v >= 0.0f ? v : 0.01f * v; }

// ---- k1: channel-summed 3x3 avg pool (zero padded), P[b,h,w] = sum_c pooled ----
__global__ void k_pool(const float* __restrict__ x, float* __restrict__ P) {
    int idx = blockIdx.x * 256 + threadIdx.x;      // 16*65536 threads
    int b = idx >> 16, hw = idx & 65535, h = hw >> 8, w = hw & 255;
    float s = 0.0f;
    for (int c = 0; c < 3; ++c) {
        const float* xc = x + ((size_t)(b * 3 + c) << 16);
        #pragma unroll
        for (int dh = -1; dh <= 1; ++dh) {
            int hh = h + dh; if ((unsigned)hh >= 256u) continue;
            #pragma unroll
            for (int dw = -1; dw <= 1; ++dw) {
                int ww = w + dw; if ((unsigned)ww >= 256u) continue;
                s += xc[(hh << 8) + ww];
            }
        }
    }
    P[idx] = s * (1.0f / 9.0f);
}

// ---- k2: LBP differences + clamp -> (16,8,256,256) ----
__global__ void k_lbp(const float* __restrict__ P, const float* __restrict__ cl,
                      const float* __restrict__ ch, float* __restrict__ lbp) {
    int idx = blockIdx.x * 256 + threadIdx.x;
    int b = idx >> 16, hw = idx & 65535, h = hw >> 8, w = hw & 255;
    float lo = cl[0], hi = ch[0];
    const float* Pb = P + ((size_t)b << 16);
    float center = Pb[hw];
    const int dh8[8] = {-1, 1, -1, 1, -1, 1, 0, 0};
    const int dw8[8] = {-1, 1,  0, 0,  1,-1, 1,-1};
    #pragma unroll
    for (int k = 0; k < 8; ++k) {
        int hh = h + dh8[k], ww = w + dw8[k];
        float nb = ((unsigned)hh < 256u && (unsigned)ww < 256u) ? Pb[(hh << 8) + ww] : 0.0f;
        float v = center - nb;
        v = fminf(fmaxf(v, lo), hi);
        lbp[(((size_t)(b * 8 + k)) << 16) + hw] = v;
    }
}

// ---- k3: per-plane min/max (128 planes of 65536) ----
__global__ void k_minmax(const float* __restrict__ lbp, float* __restrict__ MN,
                         float* __restrict__ MX) {
    int p = blockIdx.x;
    const float* v = lbp + ((size_t)p << 16);
    float mn = 3.4e38f, mx = -3.4e38f;
    for (int i = threadIdx.x; i < 65536; i += 256) {
        float t = v[i]; mn = fminf(mn, t); mx = fmaxf(mx, t);
    }
    __shared__ float smn[256], smx[256];
    int t = threadIdx.x;
    smn[t] = mn; smx[t] = mx; __syncthreads();
    for (int s = 128; s > 0; s >>= 1) {
        if (t < s) { smn[t] = fminf(smn[t], smn[t + s]); smx[t] = fmaxf(smx[t], smx[t + s]); }
        __syncthreads();
    }
    if (t == 0) { MN[p] = smn[0]; MX[p] = smx[0]; }
}

// ---- k4: joint 8x8 histogram per group (64 groups), + global sum ----
__global__ void k_hist(const float* __restrict__ lbp, const float* __restrict__ MN,
                       const float* __restrict__ MX, float* __restrict__ HIST,
                       float* __restrict__ TOT) {
    int g = blockIdx.x;                      // b4 index 0..63
    const float* v0p = lbp + ((size_t)(2 * g) << 16);
    const float* v1p = lbp + ((size_t)(2 * g + 1) << 16);
    float mn0 = MN[2 * g], mx0 = MX[2 * g], mn1 = MN[2 * g + 1], mx1 = MX[2 * g + 1];
    float r0 = mx0 - mn0, r1 = mx1 - mn1;
    float lev0[8], hi0[8], lev1[8], hi1[8];
    #pragma unroll
    for (int i = 0; i < 8; ++i) {
        lev0[i] = mn0 + r0 * (i * 0.125f);
        lev1[i] = mn1 + r1 * (i * 0.125f);
    }
    #pragma unroll
    for (int i = 0; i < 7; ++i) { hi0[i] = lev0[i + 1]; hi1[i] = lev1[i + 1]; }
    hi0[7] = mx0; hi1[7] = mx1;

    __shared__ float sh[64];
    if (threadIdx.x < 64) sh[threadIdx.x] = 0.0f;
    __syncthreads();

    for (int p = threadIdx.x; p < 65536; p += 256) {
        float v0 = v0p[p], v1 = v1p[p];
        unsigned m0 = 0, m1 = 0;
        #pragma unroll
        for (int i = 0; i < 8; ++i) {
            if (v0 >= lev0[i] && v0 <= hi0[i]) m0 |= (1u << i);
            if (v1 >= lev1[i] && v1 <= hi1[i]) m1 |= (1u << i);
        }
        float prod = v0 * v1;
        if (m0 && m1 && prod != 0.0f) {
            unsigned t0 = m0;
            while (t0) {
                int i = __ffs(t0) - 1; t0 &= t0 - 1;
                unsigned t1 = m1;
                while (t1) {
                    int j = __ffs(t1) - 1; t1 &= t1 - 1;
                    atomicAdd(&sh[i * 8 + j], prod);
                }
            }
        }
    }
    __syncthreads();
    if (threadIdx.x < 64) HIST[g * 64 + threadIdx.x] = sh[threadIdx.x];
    if (threadIdx.x == 0) {
        float s = 0.0f;
        for (int t = 0; t < 64; ++t) s += sh[t];
        atomicAdd(TOT, s);
    }
}

// ---- k5: expand (56x3 matvec) + leaky + weighted instance-norm stats on small grid ----
__global__ void k_expand(const float* __restrict__ HIST, const float* __restrict__ TOT,
                         const float* __restrict__ we, const float* __restrict__ be,
                         float* __restrict__ S, float* __restrict__ MEAN,
                         float* __restrict__ ISTD) {
    int bg = blockIdx.x;                      // b*4+g, 0..63
    __shared__ float hn[64];
    __shared__ float rs[256], rq[256];
    float tot = TOT[0];
    if (threadIdx.x < 64) hn[threadIdx.x] = HIST[bg * 64 + threadIdx.x] / tot;
    __syncthreads();

    float wsum = 0.0f, wsq = 0.0f;
    for (int idx = threadIdx.x; idx < 3584; idx += 256) {   // p in [0,64), o in [0,56)
        int p = idx / 56, o = idx - p * 56;
        float f0 = (float)(p >> 3), f1 = (float)(p & 7), f2 = hn[p];
        float val = we[o * 3 + 0] * f0 + we[o * 3 + 1] * f1 + we[o * 3 + 2] * f2 + be[o];
        val = leaky(val);
        S[(size_t)bg * 3584 + idx] = val;
        int cnt = ((o + 1) * 32 + 6) / 7 - (o * 32 + 6) / 7;  // col replication 4 or 5
        float wgt = 4.0f * (float)cnt;                        // rows replicate x4
        wsum += wgt * val;
        wsq  += wgt * val * val;
    }
    rs[threadIdx.x] = wsum; rq[threadIdx.x] = wsq; __syncthreads();
    for (int s = 128; s > 0; s >>= 1) {
        if (threadIdx.x < s) { rs[threadIdx.x] += rs[threadIdx.x + s]; rq[threadIdx.x] += rq[threadIdx.x + s]; }
        __syncthreads();
    }
    if (threadIdx.x == 0) {
        float mean = rs[0] * (1.0f / 65536.0f);
        float var  = rq[0] * (1.0f / 65536.0f) - mean * mean;
        MEAN[bg] = mean;
        ISTD[bg] = rsqrtf(var + 1e-5f);
    }
}

// ---- k6: 1x1 conv 4->16 + bias + leaky via V_WMMA_F32_16X16X4_F32 ----
// A (16x4) = w_conv, B (4x16) = 16 normalized pixels, C = bias broadcast.
__global__ void k_conv1x1_wmma(const float* __restrict__ S, const float* __restrict__ MEAN,
                               const float* __restrict__ ISTD, const float* __restrict__ wconv,
                               const float* __restrict__ bconv, float* __restrict__ Z) {
    int wave = (blockIdx.x * 256 + threadIdx.x) >> 5;   // 0..3583 (16 batches * 224 chunks)
    int lane = threadIdx.x & 31;
    int b = wave / 224;
    int chunk = wave - b * 224;
    int pix = chunk * 16 + (lane & 15);
    bool hi = lane >= 16;
    int mbase = hi ? 8 : 0;
    int l = lane & 15;

    // A-matrix 16x4 f32 layout: lanes 0-15 hold K=0,1 of row M=lane; lanes 16-31 hold K=2,3.
    v2f a;
    a.x = wconv[l * 4 + (hi ? 2 : 0)];
    a.y = wconv[l * 4 + (hi ? 3 : 1)];

    // B-matrix 4x16: lanes 0-15 hold column N=lane rows K=0,1; lanes 16-31 rows K=2,3.
    int bg0 = b * 4 + (hi ? 2 : 0);
    int bg1 = bg0 + 1;
    v2f bb;
    bb.x = (S[(size_t)bg0 * 3584 + pix] - MEAN[bg0]) * ISTD[bg0];
    bb.y = (S[(size_t)bg1 * 3584 + pix] - MEAN[bg1]) * ISTD[bg1];

    // C/D: VGPR r holds row M=r (lanes 0-15) / M=r+8 (lanes 16-31), N=lane%16.
    v8f c;
    #pragma unroll
    for (int r = 0; r < 8; ++r) c[r] = bconv[mbase + r];

    v8f d = __builtin_amdgcn_wmma_f32_16x16x4_f32(false, a, false, bb, (short)0, c, false, false);

    #pragma unroll
    for (int r = 0; r < 8; ++r) {
        float v = leaky(d[r]);
        Z[((size_t)(b * 16 + mbase + r)) * 3584 + pix] = v;
    }
}

// ---- k7: final 3x3 conv (16->1, pad 1) on the upsampled grid via index math ----
__global__ void k_final(const float* __restrict__ Z, const float* __restrict__ wf_g,
                        const float* __restrict__ bf, float* __restrict__ out) {
    __shared__ float wf[144];
    if (threadIdx.x < 144) wf[threadIdx.x] = wf_g[threadIdx.x];
    __syncthreads();

    int idx = blockIdx.x * 256 + threadIdx.x;   // 16*65536
    int b = idx >> 16, hw = idx & 65535, h = hw >> 8, w = hw & 255;
    const float* zb = Z + (size_t)b * 16 * 3584;
    float acc = bf[0];
    #pragma unroll
    for (int di = 0; di < 3; ++di) {
        int hh = h + di - 1;
        if ((unsigned)hh >= 256u) continue;
        int pr = (hh >> 2) * 56;                 // upsample rows: repeat x4
        #pragma unroll
        for (int dj = 0; dj < 3; ++dj) {
            int ww = w + dj - 1;
            if ((unsigned)ww >= 256u) continue;
            int col = (ww * 7) >> 5;             // floor(ww*56/256)
            const float* zp = zb + pr + col;
            #pragma unroll
            for (int c = 0; c < 16; ++c)
                acc += wf[c * 9 + di * 3 + dj] * zp[(size_t)c * 3584];
        }
    }
    out[idx] = leaky(acc);
}

extern "C" void kernel_launch(void* const* d_in, const int* in_sizes, int n_in,
                              void* d_out, int out_size, void* d_ws, size_t ws_size,
                              hipStream_t stream) {
    const float* x        = (const float*)d_in[0];
    const float* w_expand = (const float*)d_in[1];
    const float* b_expand = (const float*)d_in[2];
    const float* w_conv   = (const float*)d_in[3];
    const float* b_conv   = (const float*)d_in[4];
    const float* w_final  = (const float*)d_in[5];
    const float* b_final  = (const float*)d_in[6];
    const float* clamp_lo = (const float*)d_in[7];
    const float* clamp_hi = (const float*)d_in[8];

    float* ws   = (float*)d_ws;
    float* P    = ws + OFF_P;
    float* LBP  = ws + OFF_LBP;
    float* MN   = ws + OFF_MN;
    float* MX   = ws + OFF_MX;
    float* HIST = ws + OFF_HIST;
    float* TOT  = ws + OFF_TOT;
    float* S    = ws + OFF_S;
    float* MEAN = ws + OFF_MEAN;
    float* ISTD = ws + OFF_ISTD;
    float* Z    = ws + OFF_Z;

    hipMemsetAsync(TOT, 0, sizeof(float), stream);

    k_pool        <<<4096, 256, 0, stream>>>(x, P);
    k_lbp         <<<4096, 256, 0, stream>>>(P, clamp_lo, clamp_hi, LBP);
    k_minmax      <<<128,  256, 0, stream>>>(LBP, MN, MX);
    k_hist        <<<64,   256, 0, stream>>>(LBP, MN, MX, HIST, TOT);
    k_expand      <<<64,   256, 0, stream>>>(HIST, TOT, w_expand, b_expand, S, MEAN, ISTD);
    k_conv1x1_wmma<<<448,  256, 0, stream>>>(S, MEAN, ISTD, w_conv, b_conv, Z);
    k_final       <<<4096, 256, 0, stream>>>(Z, w_final, b_final, (float*)d_out);
}